// LocalStructureAnalyse_76304388980739
// MI455X (gfx1250) — compile-verified
//
#include <hip/hip_runtime.h>
#include <cstdint>
#include <cstddef>

// Structure-tensor analysis, fused single pass.
// x: (16,1,1536,1536) f32.  Outputs: 6 maps of (16,1536,1536) f32 concatenated.

constexpr int IMG_H = 1536;
constexpr int IMG_W = 1536;
constexpr int TH = 32;
constexpr int TW = 32;
constexpr int XS_C = 40;   // x-tile row stride: global cols j0-4 .. j0+35 (16B-aligned chunks)
constexpr int Q_C  = 36;   // q-tile row stride: cols j0-1 .. j0+32 used, padded to 36 (16B rows)

typedef float v4f __attribute__((ext_vector_type(4)));

__global__ __launch_bounds__(256)
void LocalStructureAnalyse_76304388980739_kernel(const float* __restrict__ x,
                                                 const float* __restrict__ weight,
                                                 float* __restrict__ out)
{
    __shared__ alignas(16) float xs [35 * XS_C];  // rows i0-2..i0+32
    __shared__ alignas(16) float qxx[34 * Q_C];   // rows i0-1..i0+32
    __shared__ alignas(16) float qxy[34 * Q_C];
    __shared__ alignas(16) float qyy[34 * Q_C];

    const int tid = threadIdx.x;
    const int i0  = blockIdx.y * TH;
    const int j0  = blockIdx.x * TW;
    const int n   = blockIdx.z;
    const float w9 = weight[0];        // 1/9 (uniform box weight)

    const float* xn = x + (size_t)n * IMG_H * IMG_W;

    // ---- Phase 1: async-copy x tile into LDS as 16B chunks (35 rows x 10 chunks) ----
    // Chunks are fully inside or fully outside the image; outside chunks get a clamped
    // (safe) address and hold garbage that is provably never consumed.
    for (int e = tid; e < 35 * 10; e += 256) {
        int lr = e / 10;
        int k  = e - lr * 10;
        int gr = i0 - 2 + lr;
        gr = gr < 0 ? 0 : (gr >= IMG_H ? IMG_H - 1 : gr);
        int gcb = j0 - 4 + 4 * k;
        gcb = gcb < 0 ? 0 : (gcb > IMG_W - 4 ? IMG_W - 4 : gcb);
        const float* src = xn + (size_t)gr * IMG_W + gcb;
        // low 32 bits of a generic pointer to LDS == wave-relative LDS byte address
        uint32_t lds_addr = (uint32_t)(uintptr_t)&xs[lr * XS_C + 4 * k];
        uint64_t gaddr    = (uint64_t)(uintptr_t)src;
        asm volatile("global_load_async_to_lds_b128 %0, %1, off"
                     :: "v"(lds_addr), "v"(gaddr) : "memory");
    }
    asm volatile("s_wait_asynccnt 0" ::: "memory");
    __syncthreads();

    // ---- Phase 2: gradients + products into LDS, vectorized 4-wide writes ----
    // q == 0 outside the image (realizes conv2d zero padding). Padding cols 34..35
    // are computed harmlessly and never used by phase 3 sums.
    for (int e = tid; e < 34 * 9; e += 256) {
        int qr = e / 9;
        int kc = e - qr * 9;
        int r  = i0 - 1 + qr;
        bool rin = (r >= 0) && (r < IMG_H);
        v4f vxx = {0.f, 0.f, 0.f, 0.f};
        v4f vxy = {0.f, 0.f, 0.f, 0.f};
        v4f vyy = {0.f, 0.f, 0.f, 0.f};
#pragma unroll
        for (int j = 0; j < 4; ++j) {
            int qc = 4 * kc + j;
            int c  = j0 - 1 + qc;
            if (rin && c >= 0 && c < IMG_W) {
                int xr = qr + 1;
                int xc = qc + 3;                        // xs col for global col c
                float x0 = xs[xr * XS_C + xc];
                int xlc = (c == 0) ? xc + 1 : xc - 1;   // col 0 uses x[.,1]
                int xur = (r == 0) ? xr + 1 : xr - 1;   // row 0 uses x[1,.]
                float gx = xs[xr * XS_C + xlc] - x0;
                float gy = xs[xur * XS_C + xc] - x0;
                vxx[j] = gx * gx; vxy[j] = gx * gy; vyy[j] = gy * gy;
            }
        }
        *(v4f*)&qxx[qr * Q_C + 4 * kc] = vxx;
        *(v4f*)&qxy[qr * Q_C + 4 * kc] = vxy;
        *(v4f*)&qyy[qr * Q_C + 4 * kc] = vyy;
    }
    __syncthreads();

    // ---- Phase 3: separable 3x3 box sums (b128 LDS reads) + eig, 4 pixels/thread ----
    const int tx = tid & 7;     // column chunk: 4 consecutive output cols
    const int ty = tid >> 3;    // output row within tile (0..31)

    float Axx[4], Axy[4], Ayy[4];
    auto filt = [&](const float* q, float* A) {
        const v4f* qv = (const v4f*)q;                  // Q_C/4 == 9 vectors per row
        v4f a = qv[(ty    ) * 9 + tx] + qv[(ty + 1) * 9 + tx] + qv[(ty + 2) * 9 + tx];
        v4f b = qv[(ty    ) * 9 + tx + 1] + qv[(ty + 1) * 9 + tx + 1] + qv[(ty + 2) * 9 + tx + 1];
        A[0] = (a[0] + a[1] + a[2]) * w9;
        A[1] = (a[1] + a[2] + a[3]) * w9;
        A[2] = (a[2] + a[3] + b[0]) * w9;
        A[3] = (a[3] + b[0] + b[1]) * w9;
    };
    filt(qxx, Axx);
    filt(qxy, Axy);
    filt(qyy, Ayy);

    v4f o1, o2, oe11, oe12, oe21, oe22;
#pragma unroll
    for (int i = 0; i < 4; ++i) {
        float axx = Axx[i], axy = Axy[i], ayy = Ayy[i];
        float T  = axx + ayy;
        float Dd = axx * ayy - axy * axy;
        float s  = sqrtf(fabsf(T * T * 0.25f - Dd));
        float l1 = T * 0.5f + s;
        float l2 = T * 0.5f - s;
        float e11 = l1 - ayy, e12 = axy;
        float e21 = l2 - ayy, e22 = axy;
        bool small = fabsf(axy) < 1e-9f;
        if (small) { e11 = 1.f; e12 = 0.f; e21 = 0.f; e22 = 1.f; }
        bool lt = fabsf(l1) < fabsf(l2);
        float L1  = lt ? l1  : l2;
        float L2  = lt ? l2  : l1;
        float E11 = lt ? e11 : e21;
        float E12 = lt ? e12 : e22;
        float E21 = lt ? e21 : e11;
        float E22 = lt ? e22 : e12;
        L1 = fmaxf(L1, 0.f);     // reference applies ReLU only to lambda1
        o1[i] = L1;  o2[i] = L2;
        oe11[i] = E11; oe12[i] = E12; oe21[i] = E21; oe22[i] = E22;
    }

    const size_t NHW  = (size_t)gridDim.z * IMG_H * IMG_W;
    const size_t base = ((size_t)n * IMG_H + (i0 + ty)) * IMG_W + (j0 + 4 * tx);
    __builtin_nontemporal_store(o1,   (v4f*)(out + 0 * NHW + base));
    __builtin_nontemporal_store(o2,   (v4f*)(out + 1 * NHW + base));
    __builtin_nontemporal_store(oe11, (v4f*)(out + 2 * NHW + base));
    __builtin_nontemporal_store(oe12, (v4f*)(out + 3 * NHW + base));
    __builtin_nontemporal_store(oe21, (v4f*)(out + 4 * NHW + base));
    __builtin_nontemporal_store(oe22, (v4f*)(out + 5 * NHW + base));
}

extern "C" void kernel_launch(void* const* d_in, const int* in_sizes, int n_in,
                              void* d_out, int out_size, void* d_ws, size_t ws_size,
                              hipStream_t stream) {
    (void)n_in; (void)out_size; (void)d_ws; (void)ws_size;
    const float* x      = (const float*)d_in[0];
    const float* weight = (const float*)d_in[1];
    float* out          = (float*)d_out;

    int N = in_sizes[0] / (IMG_H * IMG_W);   // 16
    dim3 grid(IMG_W / TW, IMG_H / TH, N);    // (48, 48, 16)
    LocalStructureAnalyse_76304388980739_kernel<<<grid, 256, 0, stream>>>(x, weight, out);
}